// OneLayerCNN_52596169507278
// MI455X (gfx1250) — compile-verified
//
#include <hip/hip_runtime.h>
#include <hip/hip_bf16.h>

typedef float v2f  __attribute__((ext_vector_type(2)));
typedef float v8f  __attribute__((ext_vector_type(8)));
typedef float v4fu __attribute__((ext_vector_type(4), aligned(4)));  // 4B-aligned b128 ld

// Conv as implicit GEMM: M = 64*129*129 pixels, N = 16 Cout, K = 48 (= 4*4*3).
// One wave -> 16x16 output tile via 12 x V_WMMA_F32_16X16X4_F32 (exact fp32).
//
// K-permutation: logical slot (s, j, khalf) -> phys k = 24*khalf + 2s + j.
// Each lane's 24 A-elements are then two 12-float contiguous runs in memory
// (input rows ih0+2*khalf, ih0+2*khalf+1), loadable as 6 x global_load_b128,
// and the loaded quads are exactly the even-aligned VGPR pairs WMMA consumes.
__global__ void __launch_bounds__(128)
conv2d_wmma_f32_kernel(const float* __restrict__ in,   // (64,256,256,3) NHWC
                       const float* __restrict__ wt,   // (4,4,3,16) HWIO: flat[k*16+co]
                       const float* __restrict__ bias, // (16)
                       float* __restrict__ out,        // (M,16) flattened
                       int tiles)
{
    const int lane  = threadIdx.x & 31;
    const int wave  = threadIdx.x >> 5;
    const int t     = blockIdx.x * (blockDim.x >> 5) + wave;
    if (t >= tiles) return;                 // wave-uniform: EXEC stays all-1s

    const int mrow  = lane & 15;
    const int khalf = lane >> 4;            // selects K half (permuted)
    const int ncol  = mrow;
    const int m0    = t << 4;
    const int m     = m0 + mrow;

    // m = n*16641 + oh*129 + ow
    const unsigned um   = (unsigned)m;
    const unsigned nimg = um / 16641u;
    const unsigned rem  = um - nimg * 16641u;
    const int oh = (int)(rem / 129u);
    const int ow = (int)(rem - (unsigned)(oh * 129));

    const float* inbase = in + (size_t)nimg * (256u * 256u * 3u);
    const int ih0 = oh * 2 - 2;             // stride 2, pad 2
    const int iw0 = ow * 2 - 2;
    const int ihb = ih0 + 2 * khalf;        // first of this lane's two input rows

    const float bval = bias[ncol];

    // B fragments under the same K-permutation: wf[s] holds phys rows
    // {24*khalf + 2s, 24*khalf + 2s + 1}, column ncol.  W flat index = k*16+co.
    v2f wf[12];
#pragma unroll
    for (int s = 0; s < 12; ++s) {
        const int k0 = 24 * khalf + 2 * s;
        wf[s].x = wt[(k0 + 0) * 16 + ncol];
        wf[s].y = wt[(k0 + 1) * 16 + ncol];
    }

    // Lane-interior test: whole 4x4 window (rows ih0..ih0+3, cols iw0..iw0+3)
    // inside the image for every lane -> pure vector-load path.
    const bool interior = (ih0 >= 0) & (ih0 + 3 < 256) & (iw0 >= 0) & (iw0 + 3 < 256);

    v2f afrag[12];

    if (__all(interior)) {
        // Fast path (~86% of tiles): 6 x b128 per lane, no masking.
        const float* r0 = inbase + ((size_t)ihb * 256 + iw0) * 3;  // 12 floats
        const float* r1 = r0 + 256 * 3;                            // next row
        v4fu q[6];
        q[0] = *(const v4fu*)(r0 + 0);
        q[1] = *(const v4fu*)(r0 + 4);
        q[2] = *(const v4fu*)(r0 + 8);
        q[3] = *(const v4fu*)(r1 + 0);
        q[4] = *(const v4fu*)(r1 + 4);
        q[5] = *(const v4fu*)(r1 + 8);
#pragma unroll
        for (int s = 0; s < 12; ++s) {
            afrag[s].x = q[s >> 1][2 * (s & 1) + 0];
            afrag[s].y = q[s >> 1][2 * (s & 1) + 1];
        }
    } else {
        // Edge path: branch-free masked scalar loads.  With the permutation,
        // idx = 2s+j in [0,24): khc = idx/12, kw = (idx%12)/3, ci = idx%3 are
        // all compile-time constants; only kh depends on khalf via ihb.
#pragma unroll
        for (int s = 0; s < 12; ++s) {
#pragma unroll
            for (int j = 0; j < 2; ++j) {
                const int idx = 2 * s + j;
                const int khc = idx / 12;
                const int r12 = idx - khc * 12;
                const int kwc = r12 / 3;
                const int cic = r12 - kwc * 3;
                const int ih  = ihb + khc;
                const int iw  = iw0 + kwc;
                const bool ok = ((unsigned)ih < 256u) & ((unsigned)iw < 256u);
                int off = (ih * 256 + iw) * 3 + cic;
                off = ok ? off : 0;               // v_cndmask, keeps load in-bounds
                const float v = inbase[off];      // unconditional load
                afrag[s][j] = ok ? v : 0.0f;      // v_cndmask, zero padding
            }
        }
    }

    v8f acc = {0.f, 0.f, 0.f, 0.f, 0.f, 0.f, 0.f, 0.f};
#pragma unroll
    for (int s = 0; s < 12; ++s) {
        acc = __builtin_amdgcn_wmma_f32_16x16x4_f32(
            /*neg_a=*/false, afrag[s], /*neg_b=*/false, wf[s],
            /*c_mod=*/(short)0, acc, /*reuse_a=*/false, /*reuse_b=*/false);
    }

    // Bias + ReLU + store. C layout: VGPR g, lane l -> M = g + 8*(l>>4), N = l&15.
#pragma unroll
    for (int g = 0; g < 8; ++g) {
        const int prow = m0 + g + khalf * 8;
        float v = acc[g] + bval;
        v = v > 0.0f ? v : 0.0f;
        out[(size_t)prow * 16 + ncol] = v;
    }
}

extern "C" void kernel_launch(void* const* d_in, const int* in_sizes, int n_in,
                              void* d_out, int out_size, void* d_ws, size_t ws_size,
                              hipStream_t stream) {
    (void)in_sizes; (void)n_in; (void)d_ws; (void)ws_size;
    const float* A = (const float*)d_in[0];    // (64,256,256,3) f32
    const float* W = (const float*)d_in[1];    // (4,4,3,16)     f32
    const float* B = (const float*)d_in[2];    // (1,1,1,16)     f32
    float*       O = (float*)d_out;

    const int Cout  = 16;
    const int Mpix  = out_size / Cout;         // 64*129*129 = 1,065,024
    const int tiles = (Mpix + 15) / 16;        // 66,564 (exact)
    const int wavesPerBlock = 4;               // 128 threads = 4 wave32
    const int blocks = (tiles + wavesPerBlock - 1) / wavesPerBlock;

    conv2d_wmma_f32_kernel<<<blocks, wavesPerBlock * 32, 0, stream>>>(A, W, B, O, tiles);
}